// graph_edgeRecovery_43284680409689
// MI455X (gfx1250) — compile-verified
//
#include <hip/hip_runtime.h>

// ---------------------------------------------------------------------------
// graph_edgeRecovery for MI455X (gfx1250, wave32)
//   out[k] = A^(k+1) @ (node_features @ Emb),  k = 0..3
//   A = dense 8192x8192 scatter of edge_weights (last-write-wins dedup)
// Bandwidth-bound: ~1.3 GB total A traffic -> ~55us floor at 23.3 TB/s.
// Matrix op:   V_WMMA_F32_16X16X4_F32 (f32 path, matches f32 reference).
// Data mover:  TENSOR_LOAD_TO_LDS (TDM) stages 16x64-f32 A tiles into LDS,
//              double-buffered per wave, TENSORcnt-tracked. TDM pad_enable
//              (pad_interval=5 -> every 64 DW, pad_amount=3 -> +4 DW) gives a
//              68-float LDS row stride = conflict-free ds_load_b64 fragments.
// ---------------------------------------------------------------------------

typedef __attribute__((ext_vector_type(2))) float v2f;
typedef __attribute__((ext_vector_type(8))) float v8f;
typedef __attribute__((ext_vector_type(4))) unsigned int u32x4;
typedef __attribute__((ext_vector_type(8))) int i32x8;
typedef __attribute__((ext_vector_type(4))) int i32x4;

#define N_NODES 8192
#define EMBD    64
#define NIN     16
#define N_EDGE  262144
#define K_ITERS 4

#define WAVES     4            // waves per spmm block (128 threads)
#define KCHUNK    64           // K elements staged per chunk (16 WMMA steps)
#define ROWPAD    68           // LDS row stride in floats (TDM pads 64 -> 68)
#define CHUNK_FLT (16 * ROWPAD)        // 1088 floats per chunk buffer
#define WBUF_FLT  (2 * CHUNK_FLT)      // double buffer per wave

// ---- TDM staging: one descriptor moves a 16x64-f32 tile into LDS ----------
__device__ __forceinline__ void tdm_stage(unsigned lds_off, unsigned long long ga) {
    // wave-uniform scalars for the SGPR-resident descriptor
    unsigned l     = __builtin_amdgcn_readfirstlane(lds_off);
    unsigned ga_lo = __builtin_amdgcn_readfirstlane((unsigned)ga);
    unsigned ga_hi = __builtin_amdgcn_readfirstlane((unsigned)(ga >> 32));

    u32x4 g0;
    g0[0] = 1u;                                  // count=1, user mode
    g0[1] = l;                                   // lds_addr (bytes)
    g0[2] = ga_lo;                               // global_addr[31:0]
    g0[3] = (ga_hi & 0x01FFFFFFu) | (2u << 30);  // global_addr[56:32] | type=2

    i32x8 g1;
    g1[0] = (2 << 16)        // data_size = 4 bytes
          | (1 << 20)        // pad_enable
          | (5 << 22)        // pad_interval: every 64 DWORDs (one row)
          | (3 << 25);       // pad_amount: 4 DWORDs  -> row stride 68 floats
    g1[1] = (int)((N_NODES & 0xFFFF) << 16);                  // tensor_dim0 lo16
    g1[2] = (int)((N_NODES >> 16) | ((N_NODES & 0xFFFF) << 16)); // dim0 hi | dim1 lo
    g1[3] = (int)((N_NODES >> 16) | (KCHUNK << 16));          // dim1 hi | tile_dim0=64
    g1[4] = 16;                                               // tile_dim1=16 rows
    g1[5] = N_NODES;                                          // dim0_stride lo32
    g1[6] = 0;                                                // stride hi | dim1_stride lo
    g1[7] = 0;

    i32x4 z4 = {0, 0, 0, 0};                 // 2-D tensor: dims 2+ unused
    i32x8 z8 = {0, 0, 0, 0, 0, 0, 0, 0};     // extra descriptor words (unused)
    __builtin_amdgcn_tensor_load_to_lds(g0, g1, z4, z4, z8, 0);
}

// ---- 1) zero the dense adjacency (256 MB) ---------------------------------
__global__ void zero_A_kernel(float4* __restrict__ p, long long n4) {
    long long i      = (long long)blockIdx.x * blockDim.x + threadIdx.x;
    long long stride = (long long)gridDim.x * blockDim.x;
    float4 z; z.x = 0.f; z.y = 0.f; z.z = 0.f; z.w = 0.f;
    for (; i < n4; i += stride) p[i] = z;
}

// ---- 2) scatter edge weights: A[r][c] = w (plain store == last-write-wins) -
__global__ void scatter_kernel(const long long* __restrict__ ei,
                               const float* __restrict__ w,
                               float* __restrict__ A) {
    int e = blockIdx.x * blockDim.x + threadIdx.x;
    if (e < N_EDGE) {
        long long r = ei[e];            // edge_index[0][e]
        long long c = ei[N_EDGE + e];   // edge_index[1][e]
        A[r * (long long)N_NODES + c] = w[e];
    }
}

// ---- 3) x = node_features @ Emb  via WMMA f32 16x16x4 ----------------------
// one wave (32 lanes) per 16-row tile; grid = 512
__global__ void embed_kernel(const float* __restrict__ nf,
                             const float* __restrict__ Emb,
                             float* __restrict__ x) {
    const int lane = threadIdx.x;          // 0..31, all active (EXEC all-1s)
    const int m    = lane & 15;            // row within tile (A) / col (B,D)
    const int kk   = (lane >> 4) * 2;      // K sub-offset per lane half
    const long long row0 = (long long)blockIdx.x * 16;

    v8f acc = {};
    for (int k = 0; k < EMBD; k += 4) {
        v2f a, b;
        const float* ap = nf + (row0 + m) * EMBD + (k + kk);
        a.x = ap[0];                        // A[m][k+kk]
        a.y = ap[1];                        // A[m][k+kk+1]
        b.x = Emb[(k + kk)     * NIN + m];  // B[k+kk][n]
        b.y = Emb[(k + kk + 1) * NIN + m];  // B[k+kk+1][n]
        acc = __builtin_amdgcn_wmma_f32_16x16x4_f32(
            /*neg_a=*/false, a, /*neg_b=*/false, b,
            /*c_mod=*/(short)0, acc, /*reuse_a=*/false, /*reuse_b=*/false);
    }
    // D layout: VGPR v -> rows v (lanes 0-15) / v+8 (lanes 16-31), N = lane&15
    const long long rbase = row0 + (lane >> 4) * 8;
    #pragma unroll
    for (int v = 0; v < 8; ++v)
        x[(rbase + v) * NIN + m] = acc[v];
}

// ---- 4) out = A @ h  ------------------------------------------------------
// one 16x16 output tile per block; 4 waves split K (2048 each);
// per-wave double-buffered TDM staging of A through LDS.
__global__ void __launch_bounds__(32 * WAVES)
spmm_kernel(const float* __restrict__ A,
            const float* __restrict__ h,
            float* __restrict__ out) {
    __shared__ float stage[WAVES * WBUF_FLT];   // 34816 B
    __shared__ float red[WAVES][256];           //  4096 B

    const int lane = threadIdx.x & 31;
    const int wave = threadIdx.x >> 5;
    const int m    = lane & 15;
    const int kk   = (lane >> 4) * 2;
    const long long row0 = (long long)blockIdx.x * 16;

    const int kspan = N_NODES / WAVES;          // 2048
    const int k0    = wave * kspan;
    const int NC    = kspan / KCHUNK;           // 32 chunks

    float* wbuf = stage + wave * WBUF_FLT;
    const float* Abase = A + row0 * (long long)N_NODES;

    v8f acc = {};

    // prefetch chunk 0  (TENSORcnt = 1)
    tdm_stage((unsigned)(size_t)wbuf,
              (unsigned long long)(size_t)(Abase + k0));

    for (int c = 0; c < NC; ++c) {
        const float* cur = wbuf + (c & 1) * CHUNK_FLT;
        const bool more = (c + 1 < NC);
        if (more)
            tdm_stage((unsigned)(size_t)(wbuf + ((c + 1) & 1) * CHUNK_FLT),
                      (unsigned long long)(size_t)(Abase + k0 + (c + 1) * KCHUNK));
        // TDM ops complete in order per wave: <=1 pending => chunk c resident
        if (more) __builtin_amdgcn_s_wait_tensorcnt((short)1);
        else      __builtin_amdgcn_s_wait_tensorcnt((short)0);
        asm volatile("" ::: "memory");           // LDS contents now visible

        const int kb = k0 + c * KCHUNK;
        const float* crow = cur + m * ROWPAD + kk;   // conflict-free ds reads
        #pragma unroll
        for (int s = 0; s < KCHUNK; s += 4) {
            v2f a = *(const v2f*)(crow + s);          // ds_load_b64 (8B aligned)
            v2f b;
            b.x = h[(kb + s + kk)     * NIN + m];     // h is L2-resident (512 KB)
            b.y = h[(kb + s + kk + 1) * NIN + m];
            acc = __builtin_amdgcn_wmma_f32_16x16x4_f32(
                false, a, false, b, (short)0, acc, false, false);
        }
    }

    // cross-wave partial-sum reduction through LDS
    #pragma unroll
    for (int v = 0; v < 8; ++v) red[wave][v * 32 + lane] = acc[v];
    __syncthreads();

    if (wave == 0) {
        #pragma unroll
        for (int v = 0; v < 8; ++v) {
            float s = 0.f;
            #pragma unroll
            for (int w2 = 0; w2 < WAVES; ++w2) s += red[w2][v * 32 + lane];
            const long long r = row0 + (lane >> 4) * 8 + v;
            out[r * NIN + m] = s;
        }
    }
}

// ---------------------------------------------------------------------------
extern "C" void kernel_launch(void* const* d_in, const int* in_sizes, int n_in,
                              void* d_out, int out_size, void* d_ws, size_t ws_size,
                              hipStream_t stream) {
    const float*     nf  = (const float*)d_in[0];      // [8192, 64] f32
    const long long* ei  = (const long long*)d_in[1];  // [2, 262144] i64
    const float*     w   = (const float*)d_in[2];      // [262144] f32
    const float*     Emb = (const float*)d_in[3];      // [64, 16] f32
    float*           out = (float*)d_out;              // [4, 8192, 16] f32

    // workspace layout: A (256 MB) | x (512 KB)
    float* A = (float*)d_ws;
    float* x = (float*)((char*)d_ws + (size_t)N_NODES * N_NODES * sizeof(float));

    // 1) A = 0
    zero_A_kernel<<<4096, 256, 0, stream>>>(
        (float4*)A, (long long)N_NODES * N_NODES / 4);

    // 2) scatter edges (dedup = last write wins)
    scatter_kernel<<<N_EDGE / 256, 256, 0, stream>>>(ei, w, A);

    // 3) x = node_features @ Emb
    embed_kernel<<<N_NODES / 16, 32, 0, stream>>>(nf, Emb, x);

    // 4) four propagation steps; slab k of d_out doubles as input to step k+1
    const float* hin = x;
    for (int k = 0; k < K_ITERS; ++k) {
        float* hout = out + (size_t)k * N_NODES * NIN;
        spmm_kernel<<<N_NODES / 16, 32 * WAVES, 0, stream>>>(A, hin, hout);
        hin = hout;
    }
}